// RGATv3Block_67851893342290
// MI455X (gfx1250) — compile-verified
//
#include <hip/hip_runtime.h>
#include <math.h>

#define NN 50000   // nodes
#define NE 800000  // edges
#define CD 128     // channels
#define NH 8       // heads
#define ED 18      // edge attr dim
#define SD 64      // seq dim

typedef __attribute__((ext_vector_type(2))) float v2f;
typedef __attribute__((ext_vector_type(8))) float v8f;

__device__ __forceinline__ float gelu_exact(float x) {
  return 0.5f * x * (1.0f + erff(x * 0.7071067811865475f));
}
// order-preserving float->uint key for atomicMax-based segment max
__device__ __forceinline__ unsigned fkey(float f) {
  unsigned b = __float_as_uint(f);
  return (b & 0x80000000u) ? ~b : (b | 0x80000000u);
}
__device__ __forceinline__ float fdec(unsigned kk) {
  unsigned b = (kk & 0x80000000u) ? (kk & 0x7fffffffu) : ~kk;
  return __uint_as_float(b);
}

// One 16x16 output tile of  A(16xKD) @ W(colsxKD)^T  with f32 WMMA, K fully
// unrolled: per step one ds_load_b64 (A pair) + one global_load_b64 (W pair)
// + one v_wmma_f32_16x16x4_f32, all with immediate offsets.
template <int KD>
__device__ __forceinline__ v8f wmma_tile(const float* __restrict__ As,
                                         const float* __restrict__ Wrow,
                                         int mrow, int ksel) {
  v8f c = {};
  const float* a_base = As + mrow * KD + ksel;
  const float* b_base = Wrow + ksel;
#pragma unroll
  for (int k0 = 0; k0 < KD; k0 += 4) {
    v2f a = *(const v2f*)(a_base + k0);
    v2f b = *(const v2f*)(b_base + k0);
    c = __builtin_amdgcn_wmma_f32_16x16x4_f32(false, a, false, b, (short)0, c,
                                              false, false);
  }
  return c;
}

// ---------------------------------------------------------------- K0: init
__global__ __launch_bounds__(256) void k0_init(float* __restrict__ acc,
                                               unsigned* __restrict__ segmax,
                                               float* __restrict__ segsum) {
  int idx = blockIdx.x * 256 + threadIdx.x;
  if (idx < NN * CD) acc[idx] = 0.0f;
  if (idx < NN * NH) { segmax[idx] = 0u; segsum[idx] = 0.0f; }
}

// ------------------------------------------- K1: LN + QKV + U via f32 WMMA
// block = 128 threads = 4 waves, handles 16 nodes.
__global__ __launch_bounds__(128) void k1_node_prep(
    const float* __restrict__ x, const float* __restrict__ seq,
    const float* __restrict__ ln_g, const float* __restrict__ ln_b,
    const float* __restrict__ Wq, const float* __restrict__ bq,
    const float* __restrict__ Wk, const float* __restrict__ bk,
    const float* __restrict__ Wv, const float* __restrict__ bv,
    const float* __restrict__ Ws1,
    float* __restrict__ q, float* __restrict__ k,
    float* __restrict__ v, float* __restrict__ u) {
  __shared__ float h_s[16][CD];   // normalized activations
  __shared__ float sn_s[16][SD];  // normalized seq features
  const int tid = threadIdx.x, lane = tid & 31;
  // uniform (SGPR) wave id: keeps tile loops + pointer selects scalar
  const int wave = __builtin_amdgcn_readfirstlane(tid >> 5);
  const int node0 = blockIdx.x * 16;

  // LayerNorm + seq L2-normalize: wave w owns nodes 4w..4w+3
#pragma unroll
  for (int i = 0; i < 4; ++i) {
    const int row = wave * 4 + i;
    const int n = node0 + row;
    float4 xv = *(const float4*)&x[(size_t)n * CD + lane * 4];
    float s1 = xv.x + xv.y + xv.z + xv.w;
    float s2 = xv.x * xv.x + xv.y * xv.y + xv.z * xv.z + xv.w * xv.w;
#pragma unroll
    for (int off = 16; off > 0; off >>= 1) {
      s1 += __shfl_xor(s1, off);
      s2 += __shfl_xor(s2, off);
    }
    float mu = s1 * (1.0f / CD);
    float inv = rsqrtf(s2 * (1.0f / CD) - mu * mu + 1e-5f);
    float4 g4 = *(const float4*)&ln_g[lane * 4];
    float4 b4 = *(const float4*)&ln_b[lane * 4];
    h_s[row][lane * 4 + 0] = (xv.x - mu) * inv * g4.x + b4.x;
    h_s[row][lane * 4 + 1] = (xv.y - mu) * inv * g4.y + b4.y;
    h_s[row][lane * 4 + 2] = (xv.z - mu) * inv * g4.z + b4.z;
    h_s[row][lane * 4 + 3] = (xv.w - mu) * inv * g4.w + b4.w;

    float2 sv = *(const float2*)&seq[(size_t)n * SD + lane * 2];
    float ss = sv.x * sv.x + sv.y * sv.y;
#pragma unroll
    for (int off = 16; off > 0; off >>= 1) ss += __shfl_xor(ss, off);
    float rn = 1.0f / fmaxf(sqrtf(ss), 1e-12f);
    sn_s[row][lane * 2 + 0] = sv.x * rn;
    sn_s[row][lane * 2 + 1] = sv.y * rn;
  }
  __syncthreads();

  // WMMA operand mapping (f32 16x16x4):
  //  A 16x4: lanes 0-15 -> M=lane, K={k0,k0+1}; lanes 16-31 -> K={k0+2,k0+3}
  //  B 4x16: lane loads contiguous pair W[col][k0+ksel .. k0+ksel+1]
  const int mrow = lane & 15;
  const int ksel = (lane >> 4) << 1;  // 0 or 2

  // q/k/v tiles: 24 tiles (mat = t>>3 in {0,1,2}, col block jb = t&7), K=128
  for (int t = wave; t < 24; t += 4) {
    const int mat = t >> 3, jb = t & 7;
    const float* Wm = (mat == 0) ? Wq : (mat == 1) ? Wk : Wv;
    const float* bm = (mat == 0) ? bq : (mat == 1) ? bk : bv;
    float* Om = (mat == 0) ? q : (mat == 1) ? k : v;
    const int col = jb * 16 + mrow;
    v8f c = wmma_tile<CD>(&h_s[0][0], Wm + (size_t)col * CD, mrow, ksel);
    const float bc = bm[col];
#pragma unroll
    for (int r = 0; r < 8; ++r) {
      const int row = (lane < 16) ? r : (r + 8);
      Om[(size_t)(node0 + row) * CD + col] = c[r] + bc;
    }
  }

  // u = sn @ Ws1^T tiles: 8 tiles, K=64, no bias (bs1 applied per edge)
  for (int jb = wave; jb < 8; jb += 4) {
    const int col = jb * 16 + mrow;
    v8f c = wmma_tile<SD>(&sn_s[0][0], Ws1 + (size_t)col * SD, mrow, ksel);
#pragma unroll
    for (int r = 0; r < 8; ++r) {
      const int row = (lane < 16) ? r : (r + 8);
      u[(size_t)(node0 + row) * CD + col] = c[r];
    }
  }
}

// --------------------------------------------- K2: per-edge logits (wave/edge)
__global__ __launch_bounds__(256) void k2_edge_logits(
    const int* __restrict__ ei, const float* __restrict__ eattr,
    const float* __restrict__ q, const float* __restrict__ k,
    const float* __restrict__ u, const float* __restrict__ bs1,
    const float* __restrict__ Ws2, const float* __restrict__ bs2,
    const float* __restrict__ We0a, const float* __restrict__ be0a,
    const float* __restrict__ We0b, const float* __restrict__ be0b,
    const float* __restrict__ We1a, const float* __restrict__ be1a,
    const float* __restrict__ We1b, const float* __restrict__ be1b,
    float* __restrict__ logits, unsigned* __restrict__ segmax) {
  __shared__ float sWs2[NH * CD];
  __shared__ float sWa[2][32 * 16];
  __shared__ float sWb[2][NH * 32];
  __shared__ float sba[2][32];
  __shared__ float sbb[2][NH];
  __shared__ float sbs1[CD];
  __shared__ float sbs2[NH];
  for (int i = threadIdx.x; i < NH * CD; i += 256) sWs2[i] = Ws2[i];
  for (int i = threadIdx.x; i < 512; i += 256) {
    sWa[0][i] = We0a[i];
    sWa[1][i] = We1a[i];
  }
  { int i = threadIdx.x;
    if (i < 256) { sWb[0][i] = We0b[i]; sWb[1][i] = We1b[i]; }
    if (i < 128) sbs1[i] = bs1[i];
    if (i < 32) { sba[0][i] = be0a[i]; sba[1][i] = be1a[i]; }
    if (i < 8) { sbb[0][i] = be0b[i]; sbb[1][i] = be1b[i]; sbs2[i] = bs2[i]; }
  }
  __syncthreads();

  const int e = blockIdx.x * 8 + (threadIdx.x >> 5);
  const int lane = threadIdx.x & 31;
  if (e >= NE) return;
  const int src = ei[e], dst = ei[NE + e];

  // per-head q.k (head h = lanes 4h..4h+3, 4 dims each)
  float4 qd = *(const float4*)&q[(size_t)dst * CD + lane * 4];
  float4 kd = *(const float4*)&k[(size_t)src * CD + lane * 4];
  float p = qd.x * kd.x + qd.y * kd.y + qd.z * kd.z + qd.w * kd.w;
  p += __shfl_xor(p, 1);
  p += __shfl_xor(p, 2);

  // seq branch: gelu(u[dst]-u[src]+bs1) dotted with Ws2 rows
  float4 ud = *(const float4*)&u[(size_t)dst * CD + lane * 4];
  float4 us = *(const float4*)&u[(size_t)src * CD + lane * 4];
  float g0 = gelu_exact(ud.x - us.x + sbs1[lane * 4 + 0]);
  float g1 = gelu_exact(ud.y - us.y + sbs1[lane * 4 + 1]);
  float g2 = gelu_exact(ud.z - us.z + sbs1[lane * 4 + 2]);
  float g3 = gelu_exact(ud.w - us.w + sbs1[lane * 4 + 3]);
  float sbp[NH];
#pragma unroll
  for (int h = 0; h < NH; ++h) {
    const float* w = &sWs2[h * CD + lane * 4];
    sbp[h] = g0 * w[0] + g1 * w[1] + g2 * w[2] + g3 * w[3];
  }

  // edge-type MLP: lane = hidden unit (32 units), ef broadcast via shfl
  float efl = (lane < ED) ? eattr[(size_t)e * ED + lane] : 0.0f;
  float a16 = __shfl(efl, 16), a17 = __shfl(efl, 17);
  const int et = (a17 > a16) ? 1 : 0;  // argmax, first index on ties
  float hacc = sba[et][lane];
#pragma unroll
  for (int i = 0; i < 16; ++i) hacc += __shfl(efl, i) * sWa[et][lane * 16 + i];
  float hj = fmaxf(hacc, 0.0f);
  float mp[NH];
#pragma unroll
  for (int h = 0; h < NH; ++h) mp[h] = hj * sWb[et][h * 32 + lane];

  float myLogit = 0.0f;
#pragma unroll
  for (int h = 0; h < NH; ++h) {
    float s = sbp[h], m = mp[h];
#pragma unroll
    for (int off = 16; off > 0; off >>= 1) {
      s += __shfl_xor(s, off);
      m += __shfl_xor(m, off);
    }
    float qkh = __shfl(p, 4 * h);
    float val = 0.25f * qkh + m + sbb[et][h] + 0.1f * tanhf(s + sbs2[h]);
    if (lane == h) myLogit = val;
  }
  if (lane < NH) {
    logits[(size_t)e * NH + lane] = myLogit;
    atomicMax(&segmax[(size_t)dst * NH + lane], fkey(myLogit));
  }
}

// ------------------------------------------ K3: exp(logit-max) + segment sum
__global__ __launch_bounds__(256) void k3_edge_exp(
    const int* __restrict__ ei, const unsigned* __restrict__ segmax,
    float* __restrict__ logits /* in: logits, out: exp */,
    float* __restrict__ segsum) {
  int idx = blockIdx.x * 256 + threadIdx.x;
  if (idx >= NE * NH) return;
  int e = idx >> 3, h = idx & 7;
  int dst = ei[NE + e];
  float mx = fdec(segmax[(size_t)dst * NH + h]);
  float ex = expf(logits[idx] - mx);
  logits[idx] = ex;
  atomicAdd(&segsum[(size_t)dst * NH + h], ex);
}

// ----------------------------------------------- K4: weighted v-aggregation
__global__ __launch_bounds__(256) void k4_aggregate(
    const int* __restrict__ ei, const float* __restrict__ ex,
    const float* __restrict__ segsum, const float* __restrict__ v,
    float* __restrict__ acc) {
  const int e = blockIdx.x * 8 + (threadIdx.x >> 5);
  if (e >= NE) return;
  const int lane = threadIdx.x & 31;
  const int src = ei[e], dst = ei[NE + e];
  const int h = lane >> 2;
  float w = ex[(size_t)e * NH + h] / (segsum[(size_t)dst * NH + h] + 1e-12f);
  float4 vv = *(const float4*)&v[(size_t)src * CD + lane * 4];
  float* a = &acc[(size_t)dst * CD + lane * 4];
  atomicAdd(a + 0, vv.x * w);
  atomicAdd(a + 1, vv.y * w);
  atomicAdd(a + 2, vv.z * w);
  atomicAdd(a + 3, vv.w * w);
}

// -------------------------------------------------- K5: gelu + residual
__global__ __launch_bounds__(256) void k5_finalize(const float* __restrict__ x,
                                                   float* __restrict__ out) {
  int idx = blockIdx.x * 256 + threadIdx.x;
  if (idx >= NN * CD) return;
  float a = out[idx];
  out[idx] = gelu_exact(a) + x[idx];
}

extern "C" void kernel_launch(void* const* d_in, const int* in_sizes, int n_in,
                              void* d_out, int out_size, void* d_ws,
                              size_t ws_size, hipStream_t stream) {
  const float* x = (const float*)d_in[0];
  const int* ei = (const int*)d_in[1];
  const float* eattr = (const float*)d_in[2];
  const float* seq = (const float*)d_in[3];
  const float* ln_g = (const float*)d_in[4];
  const float* ln_b = (const float*)d_in[5];
  const float* Wq = (const float*)d_in[6];
  const float* bq = (const float*)d_in[7];
  const float* Wk = (const float*)d_in[8];
  const float* bk = (const float*)d_in[9];
  const float* Wv = (const float*)d_in[10];
  const float* bv = (const float*)d_in[11];
  const float* We0a = (const float*)d_in[12];
  const float* be0a = (const float*)d_in[13];
  const float* We0b = (const float*)d_in[14];
  const float* be0b = (const float*)d_in[15];
  const float* We1a = (const float*)d_in[16];
  const float* be1a = (const float*)d_in[17];
  const float* We1b = (const float*)d_in[18];
  const float* be1b = (const float*)d_in[19];
  const float* Ws1 = (const float*)d_in[20];
  const float* bs1 = (const float*)d_in[21];
  const float* Ws2 = (const float*)d_in[22];
  const float* bs2 = (const float*)d_in[23];

  float* q = (float*)d_ws;
  float* k = q + (size_t)NN * CD;
  float* v = k + (size_t)NN * CD;
  float* u = v + (size_t)NN * CD;
  float* logits = u + (size_t)NN * CD;          // E*8 (reused for exp)
  unsigned* segmax = (unsigned*)(logits + (size_t)NE * NH);  // N*8
  float* segsum = (float*)(segmax + (size_t)NN * NH);        // N*8
  float* acc = (float*)d_out;  // aggregation accumulator == output buffer

  k0_init<<<(NN * CD + 255) / 256, 256, 0, stream>>>(acc, segmax, segsum);
  k1_node_prep<<<NN / 16, 128, 0, stream>>>(x, seq, ln_g, ln_b, Wq, bq, Wk, bk,
                                            Wv, bv, Ws1, q, k, v, u);
  k2_edge_logits<<<NE / 8, 256, 0, stream>>>(ei, eattr, q, k, u, bs1, Ws2, bs2,
                                             We0a, be0a, We0b, be0b, We1a, be1a,
                                             We1b, be1b, logits, segmax);
  k3_edge_exp<<<(NE * NH + 255) / 256, 256, 0, stream>>>(ei, segmax, logits,
                                                         segsum);
  k4_aggregate<<<NE / 8, 256, 0, stream>>>(ei, logits, segsum, v, acc);
  k5_finalize<<<(NN * CD + 255) / 256, 256, 0, stream>>>(x, acc);
}